// SparseCode_55645596287339
// MI455X (gfx1250) — compile-verified
//
#include <hip/hip_runtime.h>

typedef __attribute__((ext_vector_type(2))) float v2f;
typedef __attribute__((ext_vector_type(8))) float v8f;

#define T_LEN   16384
#define N_ATOMS 512
#define A_LEN   512
#define BATCH   8
#define M_TILE  32      // atoms per workgroup
#define N_TILE  256     // time positions per workgroup
#define N_ITER  32

// ---- monotonic f32 <-> u32 ordering (for packed atomic argmax) ----
__device__ __forceinline__ unsigned int f32_ord(float f) {
    unsigned int i = __float_as_uint(f);
    return (i & 0x80000000u) ? ~i : (i | 0x80000000u);
}
__device__ __forceinline__ float ord_f32(unsigned int u) {
    unsigned int i = (u & 0x80000000u) ? (u & 0x7FFFFFFFu) : ~u;
    return __uint_as_float(i);
}

// ---------------- atom normalization: dn = d / (||d|| + 1e-8) ----------------
__global__ __launch_bounds__(256) void mp_normalize(const float* __restrict__ d,
                                                    float* __restrict__ dn) {
    __shared__ float ls[8];
    const int a   = blockIdx.x;
    const int tid = threadIdx.x;
    const int wave = tid >> 5, lane = tid & 31;
    float s = 0.0f;
    for (int k = tid; k < A_LEN; k += 256) {
        float v = d[(size_t)a * A_LEN + k];
        s += v * v;
    }
    #pragma unroll
    for (int off = 16; off > 0; off >>= 1) s += __shfl_xor(s, off, 32);
    if (lane == 0) ls[wave] = s;
    __syncthreads();
    if (tid == 0) {
        float tot = 0.0f;
        #pragma unroll
        for (int w = 0; w < 8; ++w) tot += ls[w];
        ls[0] = 1.0f / (sqrtf(tot) + 1e-8f);
    }
    __syncthreads();
    const float inv = ls[0];
    for (int k = tid; k < A_LEN; k += 256)
        dn[(size_t)a * A_LEN + k] = d[(size_t)a * A_LEN + k] * inv;
}

// ---------------- init: residual = x, recon = 0, argmax keys = 0 ----------------
__global__ __launch_bounds__(256) void mp_init(const float* __restrict__ x,
                                               float* __restrict__ resid,
                                               float* __restrict__ recon,
                                               unsigned long long* __restrict__ keys) {
    int i = blockIdx.x * blockDim.x + threadIdx.x;
    if (i < BATCH * T_LEN) {
        resid[i] = x[i];
        recon[i] = 0.0f;
    }
    if (i < BATCH) keys[i] = 0ull;   // 0 == ordered encoding of -NaN => -inf sentinel
}

// ---------------- fused correlation GEMM (f32 WMMA) + global argmax ----------------
// fm[a,t] = sum_k dn[a,k] * resid[b, t+k]   (resid treated as 0 for t+k >= T)
// Never materialized: reduced to one packed (value, ~flatIdx) u64 per workgroup,
// merged with global_atomic_max_u64 -> exact jnp.argmax (first-index tie break).
__global__ __launch_bounds__(256) void mp_corr_argmax(const float* __restrict__ dn,
                                                      const float* __restrict__ resid,
                                                      unsigned long long* __restrict__ keys) {
    __shared__ float xwin[N_TILE + A_LEN];     // zero-padded residual window (3 KB)
    __shared__ unsigned long long wkeys[8];

    const int b    = blockIdx.z;
    const int tid  = threadIdx.x;
    const int wave = tid >> 5, lane = tid & 31;
    const int half = lane >> 4, l15 = lane & 15;
    const int wm   = wave >> 2, wn = wave & 3;  // 2x4 wave grid: 16 atoms x 64 times each

    // Stage residual window into LDS with zero padding past T (matches reference pad).
    const float* xb    = resid + (size_t)b * T_LEN;
    const int    tbase = blockIdx.x * N_TILE;
    for (int i = tid; i < N_TILE + A_LEN; i += 256) {
        int g = tbase + i;
        xwin[i] = (g < T_LEN) ? xb[g] : 0.0f;
    }
    __syncthreads();

    // A-fragment (16x4 f32, ISA 7.12.2): lane l -> row a0+(l&15), K = k0 + 2*(l>>4) + {0,1}
    const int    a0   = blockIdx.y * M_TILE + wm * 16;
    const float* Arow = dn + (size_t)(a0 + l15) * A_LEN + 2 * half;
    // B-fragment (4x16 f32): lane l -> col (l&15), K = k0 + 2*(l>>4) + {0,1}
    const int bbase = wn * 64 + l15 + 2 * half;

    v8f acc0 = {}, acc1 = {}, acc2 = {}, acc3 = {};
    for (int k0 = 0; k0 < A_LEN; k0 += 4) {
        v2f a = *(const v2f*)(Arow + k0);       // 8B-aligned (k0%4==0, +2*half even)
        const float* xk = xwin + bbase + k0;
        v2f b0; b0.x = xk[0];  b0.y = xk[1];
        v2f b1; b1.x = xk[16]; b1.y = xk[17];
        v2f b2; b2.x = xk[32]; b2.y = xk[33];
        v2f b3; b3.x = xk[48]; b3.y = xk[49];
        acc0 = __builtin_amdgcn_wmma_f32_16x16x4_f32(false, a, false, b0, (short)0, acc0, false, false);
        acc1 = __builtin_amdgcn_wmma_f32_16x16x4_f32(false, a, false, b1, (short)0, acc1, false, false);
        acc2 = __builtin_amdgcn_wmma_f32_16x16x4_f32(false, a, false, b2, (short)0, acc2, false, false);
        acc3 = __builtin_amdgcn_wmma_f32_16x16x4_f32(false, a, false, b3, (short)0, acc3, false, false);
    }

    // Epilogue: per-thread argmax over the 32 held C elements.
    // C layout: vgpr r, lanes 0-15 -> (M=r, N=lane); lanes 16-31 -> (M=r+8, N=lane-16)
    float        vmax     = -__builtin_inff();
    unsigned int bestflat = 0u;
    v8f accs[4] = {acc0, acc1, acc2, acc3};
    const int tcol = tbase + wn * 64 + l15;
    #pragma unroll
    for (int j = 0; j < 4; ++j) {
        #pragma unroll
        for (int r = 0; r < 8; ++r) {
            float v = accs[j][r];
            unsigned int flat = (unsigned int)(a0 + r + 8 * half) * (unsigned int)T_LEN
                              + (unsigned int)(tcol + 16 * j);
            if (v > vmax || (v == vmax && flat < bestflat)) { vmax = v; bestflat = flat; }
        }
    }
    // Pack: high = ordered value, low = ~flat (u64-max => max value, then lowest index)
    unsigned long long key =
        ((unsigned long long)f32_ord(vmax) << 32) | (unsigned long long)(~bestflat);
    #pragma unroll
    for (int off = 16; off > 0; off >>= 1) {
        unsigned long long o = __shfl_xor(key, off, 32);
        if (o > key) key = o;
    }
    if (lane == 0) wkeys[wave] = key;
    __syncthreads();
    if (tid == 0) {
        unsigned long long m = wkeys[0];
        #pragma unroll
        for (int w = 1; w < 8; ++w) if (wkeys[w] > m) m = wkeys[w];
        atomicMax(&keys[b], m);
    }
}

// ---------------- apply selected atom: resid -= v*dn[a], recon += v*dn[a] ----------------
__global__ __launch_bounds__(512) void mp_update(const float* __restrict__ dn,
                                                 unsigned long long* __restrict__ keys,
                                                 float* __restrict__ resid,
                                                 float* __restrict__ recon) {
    const int b = blockIdx.x;
    const int j = threadIdx.x;  // 0..511
    unsigned long long key = keys[b];
    float        val  = ord_f32((unsigned int)(key >> 32));
    unsigned int flat = ~(unsigned int)(key & 0xFFFFFFFFull);
    int atom = (int)(flat >> 14);          // T = 2^14
    int time = (int)(flat & (T_LEN - 1));
    int pos  = time + j;
    int end  = min(time + A_LEN, T_LEN - 1);   // reference: end = clip(time+A, 0, T-1)
    if (pos < end) {                            // mask = pos < end  (exact reference clip)
        float u = val * dn[(size_t)atom * A_LEN + j];
        size_t o = (size_t)b * T_LEN + pos;
        resid[o] -= u;
        recon[o] += u;
    }
    __syncthreads();                 // all reads of keys[b] done before reset
    if (j == 0) keys[b] = 0ull;      // re-arm argmax sentinel for next iteration
}

extern "C" void kernel_launch(void* const* d_in, const int* in_sizes, int n_in,
                              void* d_out, int out_size, void* d_ws, size_t ws_size,
                              hipStream_t stream) {
    const float* x = (const float*)d_in[0];   // (8,1,16384)
    const float* d = (const float*)d_in[1];   // (512,1,512)

    float* resid = (float*)d_out;             // output[0]: residual, 8*16384
    float* recon = resid + BATCH * T_LEN;     // output[1]: recon,    8*16384

    float* dn = (float*)d_ws;                                             // 1 MB
    unsigned long long* keys =
        (unsigned long long*)((char*)d_ws + (size_t)N_ATOMS * A_LEN * sizeof(float)); // 64 B

    mp_normalize<<<N_ATOMS, 256, 0, stream>>>(d, dn);
    mp_init<<<(BATCH * T_LEN + 255) / 256, 256, 0, stream>>>(x, resid, recon, keys);

    dim3 grid(T_LEN / N_TILE, N_ATOMS / M_TILE, BATCH);   // 64 x 16 x 8
    for (int it = 0; it < N_ITER; ++it) {
        mp_corr_argmax<<<grid, 256, 0, stream>>>(dn, resid, keys);
        mp_update<<<BATCH, A_LEN, 0, stream>>>(dn, keys, resid, recon);
    }
}